// GCN_38714835206720
// MI455X (gfx1250) — compile-verified
//
#include <hip/hip_runtime.h>

typedef __attribute__((ext_vector_type(2))) float v2f;
typedef __attribute__((ext_vector_type(8))) float v8f;

#define GCN_D 128

// ---------------------------------------------------------------------------
// WMMA GEMM: C[nrows x 128] = A[nrows x 128] @ W[128 x 128] (+ bias)
// One wave per 16-row strip; 8 column tiles of 16 -> 8 v8f accumulators.
// K loop: 32 steps of V_WMMA_F32_16X16X4_F32.
// ---------------------------------------------------------------------------
__global__ __launch_bounds__(256) void gcn_gemm_wmma(
    const float* __restrict__ A, const float* __restrict__ W,
    float* __restrict__ C, const float* __restrict__ bias, int nrows)
{
    const int lane = threadIdx.x & 31;
    const int wave = threadIdx.x >> 5;
    const int rowBase = (blockIdx.x * 8 + wave) * 16;
    if (rowBase >= nrows) return;          // wave-uniform: EXEC stays all-ones
    const int lo = lane & 15;              // M (A) / N (B,C) index within tile
    const int hi = lane >> 4;              // K half selector

    v8f acc[8] = {};
    const float* arow = A + (size_t)(rowBase + lo) * GCN_D;

    for (int k0 = 0; k0 < GCN_D; k0 += 4) {
        // A fragment: lane holds A[m][k0+2*hi], A[m][k0+2*hi+1]
        v2f a;
        a.x = arow[k0 + 2 * hi];
        a.y = arow[k0 + 2 * hi + 1];
        const float* wp = W + (size_t)(k0 + 2 * hi) * GCN_D + lo;
#pragma unroll
        for (int t = 0; t < 8; ++t) {
            // B fragment: lane holds W[k0+2*hi][n], W[k0+2*hi+1][n], n = t*16+lo
            v2f b;
            b.x = wp[t * 16];
            b.y = wp[t * 16 + GCN_D];
            acc[t] = __builtin_amdgcn_wmma_f32_16x16x4_f32(
                false, a, false, b, (short)0, acc[t], false, false);
        }
    }

    // C layout: VGPR v, lane l -> row v + 8*hi, col lo (per 16x16 f32 D layout)
#pragma unroll
    for (int t = 0; t < 8; ++t) {
        const float bv = bias ? bias[t * 16 + lo] : 0.0f;
#pragma unroll
        for (int v = 0; v < 8; ++v) {
            C[(size_t)(rowBase + v + 8 * hi) * GCN_D + t * 16 + lo] = acc[t][v] + bv;
        }
    }
}

// ---------------------------------------------------------------------------
// Degree / norm preprocessing
// ---------------------------------------------------------------------------
__global__ void gcn_deg_init(float* deg, int n) {
    int i = blockIdx.x * blockDim.x + threadIdx.x;
    if (i < n) deg[i] = 1.0f;              // self loop
}

__global__ void gcn_deg_count(const int* __restrict__ col, float* deg, int e) {
    int i = blockIdx.x * blockDim.x + threadIdx.x;
    if (i < e) atomicAdd(&deg[col[i]], 1.0f);
}

__global__ void gcn_dinv(float* deg, int n) {
    int i = blockIdx.x * blockDim.x + threadIdx.x;
    if (i < n) deg[i] = rsqrtf(deg[i]);    // in-place: deg -> dinv
}

__global__ void gcn_norm(const int* __restrict__ row, const int* __restrict__ col,
                         const float* __restrict__ dinv, float* __restrict__ nrm, int e) {
    int i = blockIdx.x * blockDim.x + threadIdx.x;
    if (i < e) nrm[i] = dinv[row[i]] * dinv[col[i]];
}

__global__ void gcn_zero(float* __restrict__ p, size_t nfloats) {
    size_t t = (size_t)blockIdx.x * blockDim.x + threadIdx.x;
    if (t * 4 < nfloats) {
        float4 z = make_float4(0.f, 0.f, 0.f, 0.f);
        ((float4*)p)[t] = z;
    }
}

// ---------------------------------------------------------------------------
// Edge scatter: one wave per edge; each lane moves float4 (32*4 = 128 feats).
// AGG[col[e]] += H[row[e]] * norm[e]   (atomic fp32 adds, L2-resident target)
// ---------------------------------------------------------------------------
__global__ __launch_bounds__(256) void gcn_scatter(
    const float* __restrict__ H, const int* __restrict__ row,
    const int* __restrict__ col, const float* __restrict__ nrm,
    float* AGG, int e)
{
    const int lane = threadIdx.x & 31;
    const int ed = blockIdx.x * 8 + (threadIdx.x >> 5);
    if (ed >= e) return;
    const int r = row[ed];
    const int c = col[ed];
    const float w = nrm[ed];
    const float4 m = ((const float4*)(H + (size_t)r * GCN_D))[lane];
    float* dst = AGG + (size_t)c * GCN_D + lane * 4;
    atomicAdd(dst + 0, m.x * w);
    atomicAdd(dst + 1, m.y * w);
    atomicAdd(dst + 2, m.z * w);
    atomicAdd(dst + 3, m.w * w);
}

// ---------------------------------------------------------------------------
// Finalize: out = [relu]( AGG + H * dinv^2 )    (self-loop folded in; AGG may
// already contain the residual X@R). out may alias AGG (pure elementwise).
// One thread per float4 -> N*32 threads.
// ---------------------------------------------------------------------------
__global__ void gcn_finalize(const float* AGG, const float* __restrict__ H,
                             const float* __restrict__ dinv, float* out,
                             int n, int relu)
{
    int t = blockIdx.x * blockDim.x + threadIdx.x;
    if (t >= n * 32) return;
    const int i = t >> 5;
    const int j = (t & 31) * 4;
    const float d = dinv[i];
    const float d2 = d * d;
    const float4 a = *(const float4*)(AGG + (size_t)i * GCN_D + j);
    const float4 h = *(const float4*)(H + (size_t)i * GCN_D + j);
    float4 o;
    o.x = a.x + h.x * d2;
    o.y = a.y + h.y * d2;
    o.z = a.z + h.z * d2;
    o.w = a.w + h.w * d2;
    if (relu) {
        o.x = fmaxf(o.x, 0.0f);
        o.y = fmaxf(o.y, 0.0f);
        o.z = fmaxf(o.z, 0.0f);
        o.w = fmaxf(o.w, 0.0f);
    }
    *(float4*)(out + (size_t)i * GCN_D + j) = o;
}

// ---------------------------------------------------------------------------
extern "C" void kernel_launch(void* const* d_in, const int* in_sizes, int n_in,
                              void* d_out, int out_size, void* d_ws, size_t ws_size,
                              hipStream_t stream)
{
    (void)n_in; (void)out_size; (void)ws_size;

    const float* x  = (const float*)d_in[0];
    const int*   ei = (const int*)d_in[1];
    const float* W1 = (const float*)d_in[2];
    const float* R1 = (const float*)d_in[3];
    const float* W2 = (const float*)d_in[4];
    const float* R2 = (const float*)d_in[5];
    const float* W3 = (const float*)d_in[6];
    const float* Wh = (const float*)d_in[7];
    const float* bh = (const float*)d_in[8];

    const int N = in_sizes[0] / GCN_D;     // 50000
    const int E = in_sizes[1] / 2;         // 600000
    const int* row = ei;                   // edge_index[0]
    const int* col = ei + E;               // edge_index[1]

    float* ws = (float*)d_ws;
    const size_t ND = (size_t)N * GCN_D;
    float* H    = ws;                      // X @ W  (messages)
    float* B1   = ws + ND;                 // ping
    float* B2   = ws + 2 * ND;             // pong
    float* deg  = ws + 3 * ND;             // deg -> dinv (N floats)
    float* nrm  = deg + N;                 // per-edge norm (E floats)

    const dim3 blk(256);
    const int gGemm = (((N + 15) / 16) + 7) / 8;           // 8 waves/block
    const int gEdge = (E + 7) / 8;                          // 8 edges/block
    const int gN    = (N + 255) / 256;
    const int gE    = (E + 255) / 256;
    const int gFin  = (N * 32 + 255) / 256;
    const int gZero = (int)((ND / 4 + 255) / 256);

    // --- preprocessing: deg (incl. self loop), dinv, per-edge norm ----------
    gcn_deg_init <<<gN, blk, 0, stream>>>(deg, N);
    gcn_deg_count<<<gE, blk, 0, stream>>>(col, deg, E);
    gcn_dinv     <<<gN, blk, 0, stream>>>(deg, N);
    gcn_norm     <<<gE, blk, 0, stream>>>(row, col, deg, nrm, E);

    // --- layer 1: relu( scatter(X@W1) + selfloop + X@R1 ) -------------------
    gcn_gemm_wmma<<<gGemm, blk, 0, stream>>>(x, W1, H, nullptr, N);
    gcn_gemm_wmma<<<gGemm, blk, 0, stream>>>(x, R1, B1, nullptr, N);   // AGG = X@R1
    gcn_scatter  <<<gEdge, blk, 0, stream>>>(H, row, col, nrm, B1, E);
    gcn_finalize <<<gFin, blk, 0, stream>>>(B1, H, deg, B1, N, 1);

    // --- layer 2 ------------------------------------------------------------
    gcn_gemm_wmma<<<gGemm, blk, 0, stream>>>(B1, W2, H, nullptr, N);
    gcn_gemm_wmma<<<gGemm, blk, 0, stream>>>(B1, R2, B2, nullptr, N);  // AGG = X@R2
    gcn_scatter  <<<gEdge, blk, 0, stream>>>(H, row, col, nrm, B2, E);
    gcn_finalize <<<gFin, blk, 0, stream>>>(B2, H, deg, B2, N, 1);

    // --- layer 3: no residual, no activation --------------------------------
    gcn_gemm_wmma<<<gGemm, blk, 0, stream>>>(B2, W3, H, nullptr, N);
    gcn_zero     <<<gZero, blk, 0, stream>>>(B1, ND);
    gcn_scatter  <<<gEdge, blk, 0, stream>>>(H, row, col, nrm, B1, E);
    gcn_finalize <<<gFin, blk, 0, stream>>>(B1, H, deg, B1, N, 0);

    // --- head: out = X @ Wh + bh -------------------------------------------
    gcn_gemm_wmma<<<gGemm, blk, 0, stream>>>(B1, Wh, (float*)d_out, bh, N);
}